// LocalPooling_32392643346850
// MI455X (gfx1250) — compile-verified
//
#include <hip/hip_runtime.h>
#include <hip/hip_bf16.h>
#include <math.h>

// ---------------------------------------------------------------------------
// ASAP-style pooling pipeline. Problem constants derived from in_sizes:
//   C = 128, N = 2048, E = N*128 + N = 264192, K = N/2 = 1024
// Outputs (flat, float32): x_new [K*C] | A_coarse [K*K] | S [N*K] | perm [K]
//
// GEMMs (x@W, A@S, S^T@(AS), ~13 GFLOP total) run on V_WMMA_F32_16X16X4_F32:
// fp32 in/out matches the fp32 reference exactly, and at these sizes the
// pipeline is L2/atomic bound, not matrix-ALU bound, so there is no reason to
// drop precision for a faster WMMA shape.
// ---------------------------------------------------------------------------

typedef float v2f __attribute__((ext_vector_type(2)));
typedef float v8f __attribute__((ext_vector_type(8)));

// ------------------------- generic helpers ---------------------------------

__global__ void zero_kernel(float* p, long n) {
  long i = (long)blockIdx.x * blockDim.x + threadIdx.x;
  if (i < n) p[i] = 0.0f;
}

__device__ __forceinline__ void atomicMaxF(float* addr, float val) {
  // ordered-float atomic max: positives compare as int, negatives as uint
  if (val >= 0.0f) atomicMax((int*)addr, __float_as_int(val));
  else             atomicMin((unsigned int*)addr, (unsigned int)__float_as_int(val));
}

// ------------------------- FP32 WMMA GEMM ----------------------------------
// C[M,N] = A * B   (TRANSA: A_op[m][k] = A[k*lda + m])
// One 32x32 macro-tile per wave = 2x2 WMMA 16x16 subtiles; each A/B fragment
// feeds two v_wmma_f32_16x16x4_f32 ops (2x operand reuse). K advances in
// panels of 32 with the 8 inner steps fully unrolled so the scheduler can
// clause the loads; the next panel is prefetched (global_prefetch_b8).
// Full blocks, wave-uniform control flow => EXEC all-1s at every WMMA.
template <int TRANSA>
__global__ void wmma_gemm_f32(const float* __restrict__ A,
                              const float* __restrict__ B,
                              float* __restrict__ C,
                              int M, int Nc, int Kd,
                              int lda, int ldb, int ldc) {
  const int wavesPerBlock = blockDim.x >> 5;
  const int waveId = blockIdx.x * wavesPerBlock + (threadIdx.x >> 5);
  const int tilesN = Nc >> 5;                  // 32-wide macro tiles
  const int tilesTotal = (M >> 5) * tilesN;
  if (waveId >= tilesTotal) return;            // wave-uniform branch
  const int tileM = (waveId / tilesN) << 5;
  const int tileN = (waveId % tilesN) << 5;
  const int lane = threadIdx.x & 31;
  const int half = lane >> 4;                  // 0: lanes 0-15, 1: lanes 16-31
  const int r = lane & 15;

  v8f acc00 = {0.f,0.f,0.f,0.f,0.f,0.f,0.f,0.f};
  v8f acc01 = acc00, acc10 = acc00, acc11 = acc00;

  for (int kp = 0; kp < Kd; kp += 32) {
    // prefetch next K-panel into cache while this one computes
    if (kp + 32 < Kd) {
      const int kn = kp + 32 + 2 * half;
      __builtin_prefetch(&B[(size_t)kn * ldb + tileN + r], 0, 0);
      __builtin_prefetch(&B[(size_t)kn * ldb + tileN + 16 + r], 0, 0);
      if (TRANSA) {
        __builtin_prefetch(&A[(size_t)kn * lda + tileM + r], 0, 0);
        __builtin_prefetch(&A[(size_t)kn * lda + tileM + 16 + r], 0, 0);
      } else {
        __builtin_prefetch(&A[(size_t)(tileM + r) * lda + kp + 32], 0, 0);
        __builtin_prefetch(&A[(size_t)(tileM + 16 + r) * lda + kp + 32], 0, 0);
      }
    }
    #pragma unroll
    for (int kk = 0; kk < 32; kk += 4) {
      const int k0 = kp + kk + 2 * half;
      v2f a0, a1, b0, b1;
      if (TRANSA) {
        a0.x = A[(size_t)(k0    ) * lda + tileM + r];
        a0.y = A[(size_t)(k0 + 1) * lda + tileM + r];
        a1.x = A[(size_t)(k0    ) * lda + tileM + 16 + r];
        a1.y = A[(size_t)(k0 + 1) * lda + tileM + 16 + r];
      } else {
        a0 = *(const v2f*)&A[(size_t)(tileM + r) * lda + k0];      // 8B load
        a1 = *(const v2f*)&A[(size_t)(tileM + 16 + r) * lda + k0]; // 8B load
      }
      b0.x = B[(size_t)(k0    ) * ldb + tileN + r];
      b0.y = B[(size_t)(k0 + 1) * ldb + tileN + r];
      b1.x = B[(size_t)(k0    ) * ldb + tileN + 16 + r];
      b1.y = B[(size_t)(k0 + 1) * ldb + tileN + 16 + r];
      // (neg_a, A, neg_b, B, c_mod, C, reuse_a, reuse_b)
      acc00 = __builtin_amdgcn_wmma_f32_16x16x4_f32(false, a0, false, b0,
                                                    (short)0, acc00, false, false);
      acc01 = __builtin_amdgcn_wmma_f32_16x16x4_f32(false, a0, false, b1,
                                                    (short)0, acc01, false, false);
      acc10 = __builtin_amdgcn_wmma_f32_16x16x4_f32(false, a1, false, b0,
                                                    (short)0, acc10, false, false);
      acc11 = __builtin_amdgcn_wmma_f32_16x16x4_f32(false, a1, false, b1,
                                                    (short)0, acc11, false, false);
    }
  }
  #pragma unroll
  for (int i = 0; i < 8; ++i) {
    const int m0 = tileM + i + 8 * half;
    C[(size_t)m0 * ldc + tileN + r]            = acc00[i];
    C[(size_t)m0 * ldc + tileN + 16 + r]       = acc01[i];
    C[(size_t)(m0 + 16) * ldc + tileN + r]     = acc10[i];
    C[(size_t)(m0 + 16) * ldc + tileN + 16 + r] = acc11[i];
  }
}

// ------------------------- BatchNorm ---------------------------------------

__global__ void bn_stats(const float* __restrict__ x, float* mu, float* rstd,
                         int Nn, int Cc) {
  const int c = blockIdx.x;
  __shared__ float sh[256], sh2[256];
  float s = 0.f, s2 = 0.f;
  for (int i = threadIdx.x; i < Nn; i += blockDim.x) {
    float v = x[(size_t)i * Cc + c];
    s += v; s2 += v * v;
  }
  sh[threadIdx.x] = s; sh2[threadIdx.x] = s2; __syncthreads();
  for (int st = blockDim.x >> 1; st > 0; st >>= 1) {
    if ((int)threadIdx.x < st) { sh[threadIdx.x] += sh[threadIdx.x + st];
                                 sh2[threadIdx.x] += sh2[threadIdx.x + st]; }
    __syncthreads();
  }
  if (threadIdx.x == 0) {
    float m = sh[0] / (float)Nn;
    float var = sh2[0] / (float)Nn - m * m;
    mu[c] = m;
    rstd[c] = rsqrtf(var + 1e-5f);
  }
}

__global__ void bn_apply(const float* __restrict__ x, const float* mu,
                         const float* rstd, const float* g, const float* b,
                         float* __restrict__ xhat, long n, int Cc) {
  long i = (long)blockIdx.x * blockDim.x + threadIdx.x;
  if (i < n) {
    int c = (int)(i % Cc);
    xhat[i] = (x[i] - mu[c]) * rstd[c] * g[c] + b[c];
  }
}

// ------------------------- GCN ---------------------------------------------

__global__ void edge_deg(const int* __restrict__ col, const float* __restrict__ w,
                         float* deg, int E) {
  int e = blockIdx.x * blockDim.x + threadIdx.x;
  if (e < E) atomicAdd(&deg[col[e]], w[e]);
}

__global__ void make_dinv(const float* deg, float* dinv, int Nn) {
  int i = blockIdx.x * blockDim.x + threadIdx.x;
  if (i < Nn) dinv[i] = deg[i] > 0.0f ? rsqrtf(deg[i]) : 0.0f;
}

// one block (C threads) per edge: xg[col] += norm * h[row]
__global__ void gcn_agg(const int* __restrict__ row, const int* __restrict__ col,
                        const float* __restrict__ w, const float* __restrict__ dinv,
                        const float* __restrict__ h, float* __restrict__ xg, int Cc) {
  int e = blockIdx.x;
  int rr = row[e], cc = col[e];
  float norm = dinv[rr] * w[e] * dinv[cc];
  atomicAdd(&xg[(size_t)cc * Cc + threadIdx.x],
            norm * h[(size_t)rr * Cc + threadIdx.x]);
}

// per-node squash (optionally + bias first); in place
__global__ void squash_rows(float* __restrict__ v, const float* bias, int Cc) {
  int n = blockIdx.x;
  __shared__ float sh[256];
  float val = v[(size_t)n * Cc + threadIdx.x];
  if (bias) val += bias[threadIdx.x];
  sh[threadIdx.x] = val * val; __syncthreads();
  for (int st = blockDim.x >> 1; st > 0; st >>= 1) {
    if ((int)threadIdx.x < st) sh[threadIdx.x] += sh[threadIdx.x + st];
    __syncthreads();
  }
  float n2 = sh[0];
  float sc = (n2 / (1.0f + n2)) * rsqrtf(n2 + 1e-8f);
  v[(size_t)n * Cc + threadIdx.x] = val * sc;
}

// ------------------------- dynamic routing ---------------------------------

__global__ void copy_f(const float* src, float* dst, long n) {
  long i = (long)blockIdx.x * blockDim.x + threadIdx.x;
  if (i < n) dst[i] = src[i];
}

__global__ void route_init(float* clu, float* m, float* s, int Cc) {
  int n = blockIdx.x;
  clu[(size_t)n * Cc + threadIdx.x] = 0.0f;
  if (threadIdx.x == 0) { m[n] = -__builtin_inff(); s[n] = 0.0f; }
}

__global__ void seg_max(const float* __restrict__ bij, const int* __restrict__ col,
                        float* m, int E) {
  int e = blockIdx.x * blockDim.x + threadIdx.x;
  if (e < E) atomicMaxF(&m[col[e]], bij[e]);
}

__global__ void seg_exp_sum(const float* __restrict__ bij, const int* __restrict__ col,
                            const float* __restrict__ m, float* cij, float* s, int E) {
  int e = blockIdx.x * blockDim.x + threadIdx.x;
  if (e < E) {
    float v = expf(bij[e] - m[col[e]]);
    cij[e] = v;
    atomicAdd(&s[col[e]], v);
  }
}

// one block (C threads) per edge: normalize c_ij, clu[row] += c * xs[col]
__global__ void route_accum(const int* __restrict__ row, const int* __restrict__ col,
                            float* __restrict__ cij, const float* __restrict__ s,
                            const float* __restrict__ xs, float* __restrict__ clu,
                            int Cc) {
  int e = blockIdx.x;
  int rr = row[e], cc = col[e];
  float c = cij[e] / (s[cc] + 1e-16f);
  __syncthreads();                       // everyone read cij[e] before rewrite
  if (threadIdx.x == 0) cij[e] = c;
  atomicAdd(&clu[(size_t)rr * Cc + threadIdx.x],
            c * xs[(size_t)cc * Cc + threadIdx.x]);
}

// one block per edge: b_ij += dot(clu[row], xs[col])
__global__ void route_bupdate(const int* __restrict__ row, const int* __restrict__ col,
                              const float* __restrict__ clu, const float* __restrict__ xs,
                              float* __restrict__ bij, int Cc) {
  int e = blockIdx.x;
  int rr = row[e], cc = col[e];
  __shared__ float sh[256];
  sh[threadIdx.x] = clu[(size_t)rr * Cc + threadIdx.x] *
                    xs[(size_t)cc * Cc + threadIdx.x];
  __syncthreads();
  for (int st = blockDim.x >> 1; st > 0; st >>= 1) {
    if ((int)threadIdx.x < st) sh[threadIdx.x] += sh[threadIdx.x + st];
    __syncthreads();
  }
  if (threadIdx.x == 0) bij[e] += sh[0];
}

// ------------------------- scoring / top-k ---------------------------------

__global__ void row_norm(const float* __restrict__ v, float* score, int Cc) {
  int n = blockIdx.x;
  __shared__ float sh[256];
  float x = v[(size_t)n * Cc + threadIdx.x];
  sh[threadIdx.x] = x * x; __syncthreads();
  for (int st = blockDim.x >> 1; st > 0; st >>= 1) {
    if ((int)threadIdx.x < st) sh[threadIdx.x] += sh[threadIdx.x + st];
    __syncthreads();
  }
  if (threadIdx.x == 0) score[n] = sqrtf(sh[0]);
}

// rank of node i among descending scores (ties -> lower index first, matching
// jax.lax.top_k stability). Block per node, 256 threads scan all N.
__global__ void topk_rank(const float* __restrict__ score, int* rank, int* sel,
                          int* perm, int Nn, int Kk) {
  int i = blockIdx.x;
  float si = score[i];
  __shared__ int sh[256];
  int cnt = 0;
  for (int j = threadIdx.x; j < Nn; j += blockDim.x) {
    float sj = score[j];
    if (sj > si || (sj == si && j < i)) cnt++;
  }
  sh[threadIdx.x] = cnt; __syncthreads();
  for (int st = blockDim.x >> 1; st > 0; st >>= 1) {
    if ((int)threadIdx.x < st) sh[threadIdx.x] += sh[threadIdx.x + st];
    __syncthreads();
  }
  if (threadIdx.x == 0) {
    int rk = sh[0];
    rank[i] = rk;
    int isSel = rk < Kk;
    sel[i] = isSel;
    if (isSel) perm[rk] = i;
  }
}

// ------------------------- S / A construction ------------------------------

__global__ void scatter_S(const int* __restrict__ row, const int* __restrict__ col,
                          const float* __restrict__ cij, const int* __restrict__ sel,
                          const int* __restrict__ rank, float* S, int E, int Kk) {
  int e = blockIdx.x * blockDim.x + threadIdx.x;
  if (e < E) {
    int rr = row[e];
    if (sel[rr]) atomicAdd(&S[(size_t)col[e] * Kk + rank[rr]], cij[e]);
  }
}

__global__ void scatter_A(const int* __restrict__ row, const int* __restrict__ col,
                          const float* __restrict__ w, float* A, int E, int Nn) {
  int e = blockIdx.x * blockDim.x + threadIdx.x;
  if (e < E) atomicAdd(&A[(size_t)row[e] * Nn + col[e]], w[e]);
}

__global__ void gather_out(const float* __restrict__ clu, const int* __restrict__ perm,
                           float* __restrict__ xnew, float* __restrict__ perm_out,
                           int Cc) {
  int j = blockIdx.x;
  int p = perm[j];
  xnew[(size_t)j * Cc + threadIdx.x] = clu[(size_t)p * Cc + threadIdx.x];
  if (threadIdx.x == 0) perm_out[j] = (float)p;
}

// ---------------------------------------------------------------------------

extern "C" void kernel_launch(void* const* d_in, const int* in_sizes, int n_in,
                              void* d_out, int out_size, void* d_ws, size_t ws_size,
                              hipStream_t stream) {
  const float* x      = (const float*)d_in[0];
  const int*   ei     = (const int*)  d_in[1];
  const float* w      = (const float*)d_in[2];
  const float* gamma  = (const float*)d_in[3];
  const float* beta   = (const float*)d_in[4];
  const float* gcn_w  = (const float*)d_in[5];
  const float* gcn_b  = (const float*)d_in[6];

  const int C  = in_sizes[3];            // 128
  const int Nn = in_sizes[0] / C;        // 2048
  const int E  = in_sizes[2];            // 264192
  const int Kk = Nn / 2;                 // 1024
  const int* row = ei;
  const int* col = ei + E;

  // ---- workspace carve-up (floats) ----
  float* ws = (float*)d_ws;
  size_t off = 0;
  auto alloc = [&](size_t n) { float* p = ws + off; off += n; return p; };
  float* xhat  = alloc((size_t)Nn * C);
  float* h     = alloc((size_t)Nn * C);
  float* mu    = alloc(C);
  float* rstd  = alloc(C);
  float* deg   = alloc(Nn);
  float* dinv  = alloc(Nn);
  float* xs    = alloc((size_t)Nn * C);      // x_gcn, then squash'd in place
  float* bij   = alloc(E);
  float* cij   = alloc(E);
  float* mseg  = alloc(Nn);
  float* sseg  = alloc(Nn);
  float* clu   = alloc((size_t)Nn * C);
  float* score = alloc(Nn);
  int*   rank  = (int*)alloc(Nn);
  int*   sel   = (int*)alloc(Nn);
  int*   perm  = (int*)alloc(Kk);
  float* Amat  = alloc((size_t)Nn * Nn);
  float* AS    = alloc((size_t)Nn * Kk);

  // ---- output carve-up ----
  float* out      = (float*)d_out;
  float* xnew_o   = out;                                    // K*C
  float* Acoarse  = out + (size_t)Kk * C;                   // K*K
  float* S        = out + (size_t)Kk * C + (size_t)Kk * Kk; // N*K
  float* perm_o   = S + (size_t)Nn * Kk;                    // K

  auto zero = [&](float* p, long n) {
    zero_kernel<<<(int)((n + 255) / 256), 256, 0, stream>>>(p, n);
  };
  auto gemm = [&](const float* Ag, const float* Bg, float* Cg,
                  int M, int Nc, int Kd, int lda, int ldb, int ldc, int tA) {
    int tiles = (M / 32) * (Nc / 32);
    int blocks = (tiles + 3) / 4;            // 4 waves / 128-thread block
    if (tA)
      wmma_gemm_f32<1><<<blocks, 128, 0, stream>>>(Ag, Bg, Cg, M, Nc, Kd,
                                                   lda, ldb, ldc);
    else
      wmma_gemm_f32<0><<<blocks, 128, 0, stream>>>(Ag, Bg, Cg, M, Nc, Kd,
                                                   lda, ldb, ldc);
  };

  // ---- zero accumulators ----
  zero(deg, Nn);
  zero(xs, (long)Nn * C);
  zero(Amat, (long)Nn * Nn);
  zero(S, (long)Nn * Kk);

  // ---- BatchNorm ----
  bn_stats<<<C, 256, 0, stream>>>(x, mu, rstd, Nn, C);
  {
    long n = (long)Nn * C;
    bn_apply<<<(int)((n + 255) / 256), 256, 0, stream>>>(x, mu, rstd, gamma,
                                                         beta, xhat, n, C);
  }

  // ---- h = xhat @ W   (WMMA fp32) ----
  gemm(xhat, gcn_w, h, Nn, C, C, C, C, C, 0);

  // ---- GCN normalization + aggregation ----
  edge_deg<<<(E + 255) / 256, 256, 0, stream>>>(col, w, deg, E);
  make_dinv<<<(Nn + 255) / 256, 256, 0, stream>>>(deg, dinv, Nn);
  gcn_agg<<<E, C, 0, stream>>>(row, col, w, dinv, h, xs, C);
  squash_rows<<<Nn, C, 0, stream>>>(xs, gcn_b, C);   // xs = squash(x_gcn + b)

  // ---- dynamic routing (3 iterations, b-update after first two) ----
  copy_f<<<(E + 255) / 256, 256, 0, stream>>>(w, bij, E);
  for (int it = 0; it < 3; ++it) {
    route_init<<<Nn, C, 0, stream>>>(clu, mseg, sseg, C);
    seg_max<<<(E + 255) / 256, 256, 0, stream>>>(bij, col, mseg, E);
    seg_exp_sum<<<(E + 255) / 256, 256, 0, stream>>>(bij, col, mseg, cij, sseg, E);
    route_accum<<<E, C, 0, stream>>>(row, col, cij, sseg, xs, clu, C);
    squash_rows<<<Nn, C, 0, stream>>>(clu, nullptr, C);
    if (it < 2)
      route_bupdate<<<E, C, 0, stream>>>(row, col, clu, xs, bij, C);
  }
  // clu == cluster (squashed), cij == final c_ij

  // ---- score + top-k selection ----
  row_norm<<<Nn, C, 0, stream>>>(clu, score, C);
  topk_rank<<<Nn, 256, 0, stream>>>(score, rank, sel, perm, Nn, Kk);

  // ---- S and A ----
  scatter_S<<<(E + 255) / 256, 256, 0, stream>>>(row, col, cij, sel, rank, S, E, Kk);
  scatter_A<<<(E + 255) / 256, 256, 0, stream>>>(row, col, w, Amat, E, Nn);

  // ---- A_coarse = S^T (A S)  (WMMA fp32) ----
  gemm(Amat, S, AS, Nn, Kk, Nn, Nn, Kk, Kk, 0);        // AS = A @ S
  gemm(S, AS, Acoarse, Kk, Kk, Nn, Kk, Kk, Kk, 1);     // S^T @ AS

  // ---- x_new, perm ----
  gather_out<<<Kk, C, 0, stream>>>(clu, perm, xnew_o, perm_o, C);
}